// RangeLossOp_64123861729955
// MI455X (gfx1250) — compile-verified
//
#include <hip/hip_runtime.h>
#include <hip/hip_bf16.h>

typedef float v2f __attribute__((ext_vector_type(2)));
typedef float v8f __attribute__((ext_vector_type(8)));

#define NUM_CLASSES 1000
#define CPAD        1024      // pad classes to multiple of 32 for 2x2 WMMA macro-tiles
#define DDIM        512
#define NROWS       131072
#define MAXR        256       // binomial(131072,1/1000): mean 131, sigma ~11 -> 256 is ~11 sigma
#define INF_BITS    0x7F800000u

// workspace layout (4-byte units):
//   [0, CPAD*DDIM)                    : centers (2 MB; rows >= NUM_CLASSES zeroed each call)
//   [CPAD*DDIM, +CPAD)                : counts (int)
//   [CPAD*DDIM+CPAD, +CPAD)           : sq = ||center||^2  (+inf for padded rows)
//   [CPAD*DDIM+2*CPAD]                : global min (uint bits of positive float)
//   [CPAD*DDIM+2*CPAD+64, +CPAD*MAXR) : per-class row lists (int)
#define OFF_COUNTS  (CPAD * DDIM)
#define OFF_SQ      (CPAD * DDIM + CPAD)
#define OFF_MIN     (CPAD * DDIM + 2 * CPAD)
#define OFF_ROWS    (CPAD * DDIM + 2 * CPAD + 64)

__global__ void rl_init(float* ws) {
    int i = blockIdx.x * blockDim.x + threadIdx.x;
    if (i < (CPAD - NUM_CLASSES) * DDIM) ws[NUM_CLASSES * DDIM + i] = 0.0f;       // pad center rows
    if (i < CPAD) ((int*)(ws + OFF_COUNTS))[i] = 0;                                // counts
    if (i < CPAD - NUM_CLASSES) ws[OFF_SQ + NUM_CLASSES + i] = __int_as_float((int)INF_BITS);
    if (i == 0) ((unsigned int*)ws)[OFF_MIN] = INF_BITS;
}

// Pass 1: per-class row lists. Only 131K int atomics to 1000 counters (negligible),
// instead of 67M f32 atomics in a scatter design.
__global__ void rl_build_rows(const int* __restrict__ labels,
                              int* __restrict__ counts,
                              int* __restrict__ rowlist) {
    int row = blockIdx.x * blockDim.x + threadIdx.x;
    int lab = labels[row];
    int slot = atomicAdd(&counts[lab], 1);
    if (slot < MAXR) rowlist[lab * MAXR + slot] = row;
}

// Pass 2: one 256-thread WG per class gathers its ~131 rows (2KB contiguous each,
// fully coalesced, independent loads -> deep LOADcnt pipelining) and reduces in
// registers. Fuses mean + ||center||^2. Bitonic sort of the row list gives a
// canonical summation order -> bit-deterministic across calls.
__global__ __launch_bounds__(256)
void rl_class_reduce(const float* __restrict__ feats,
                     const int* __restrict__ counts,
                     const int* __restrict__ rowlist,
                     float* __restrict__ centers,
                     float* __restrict__ sq) {
    __shared__ int   lsRows[MAXR];
    __shared__ float lsRed[8];
    int c = blockIdx.x;
    int t = threadIdx.x;
    int cnt = counts[c];
    int n = cnt < MAXR ? cnt : MAXR;

    lsRows[t] = (t < n) ? rowlist[c * MAXR + t] : 0x7FFFFFFF;
    __syncthreads();
    for (int ksz = 2; ksz <= MAXR; ksz <<= 1) {          // bitonic sort (ints, in LDS)
        for (int j = ksz >> 1; j > 0; j >>= 1) {
            int ixj = t ^ j;
            if (ixj > t) {
                int a = lsRows[t], b = lsRows[ixj];
                bool asc = ((t & ksz) == 0);
                if ((a > b) == asc) { lsRows[t] = b; lsRows[ixj] = a; }
            }
            __syncthreads();
        }
    }

    float ax = 0.0f, ay = 0.0f;                          // thread owns cols 2t, 2t+1
    const float* colp = feats + 2 * t;
#pragma unroll 4
    for (int r = 0; r < n; ++r) {
        int row = lsRows[r];                             // LDS broadcast
        v2f v = *(const v2f*)(colp + (size_t)row * DDIM);
        ax += v.x; ay += v.y;
    }
    float inv = 1.0f / fmaxf((float)cnt, 1.0f);
    ax *= inv; ay *= inv;
    v2f cv; cv.x = ax; cv.y = ay;
    *(v2f*)(centers + (size_t)c * DDIM + 2 * t) = cv;

    float s = ax * ax + ay * ay;
    for (int off = 16; off; off >>= 1) s += __shfl_xor(s, off, 32);
    if ((t & 31) == 0) lsRed[t >> 5] = s;
    __syncthreads();
    if (t == 0) {
        float tot = 0.0f;
#pragma unroll
        for (int w = 0; w < 8; ++w) tot += lsRed[w];
        sq[c] = tot;
    }
}

// One wave per 32x32 Gram macro-tile (upper triangle only). 2x2 WMMA f32 tiles,
// K-loop of V_WMMA_F32_16X16X4_F32, fused distance + masked-min reduction.
// Padded rows carry sq=+inf so they self-mask without predicates.
__global__ __launch_bounds__(32)
void rl_gram_min(const float* __restrict__ centers,
                 const float* __restrict__ sq,
                 unsigned int* __restrict__ minbits) {
    int bx = blockIdx.x, by = blockIdx.y;
    if (bx > by) return;                // Gram is symmetric; diagonal tiles kept
    int lane = threadIdx.x;
    int half = lane >> 4;               // K phase select (lanes 16-31 -> K+2)
    int l    = lane & 15;
    int row0 = bx * 32, col0 = by * 32;

    const float* a0p = centers + (size_t)(row0 + l)      * DDIM + 2 * half;
    const float* a1p = centers + (size_t)(row0 + 16 + l) * DDIM + 2 * half;
    const float* b0p = centers + (size_t)(col0 + l)      * DDIM + 2 * half;
    const float* b1p = centers + (size_t)(col0 + 16 + l) * DDIM + 2 * half;

    v8f acc00 = {}, acc01 = {}, acc10 = {}, acc11 = {};
#pragma unroll 4
    for (int k = 0; k < DDIM; k += 4) {
        v2f a0 = *(const v2f*)(a0p + k);
        v2f a1 = *(const v2f*)(a1p + k);
        v2f b0 = *(const v2f*)(b0p + k);
        v2f b1 = *(const v2f*)(b1p + k);
        acc00 = __builtin_amdgcn_wmma_f32_16x16x4_f32(false, a0, false, b0, (short)0, acc00, false, false);
        acc01 = __builtin_amdgcn_wmma_f32_16x16x4_f32(false, a0, false, b1, (short)0, acc01, false, false);
        acc10 = __builtin_amdgcn_wmma_f32_16x16x4_f32(false, a1, false, b0, (short)0, acc10, false, false);
        acc11 = __builtin_amdgcn_wmma_f32_16x16x4_f32(false, a1, false, b1, (short)0, acc11, false, false);
    }

    float lmin = __int_as_float((int)INF_BITS);
    auto fold = [&](const v8f& acc, int i0, int j0) {
        float sj = sq[j0 + l];          // N = lane%16
#pragma unroll
        for (int v = 0; v < 8; ++v) {
            int i = i0 + v + 8 * half;  // M = vgpr + 8*(lane/16)
            float dis = sq[i] + sj - 2.0f * acc[v];
            if (dis > 0.0f && dis < lmin) lmin = dis;
        }
    };
    fold(acc00, row0,      col0);
    fold(acc01, row0,      col0 + 16);
    fold(acc10, row0 + 16, col0);
    fold(acc11, row0 + 16, col0 + 16);

    for (int off = 16; off; off >>= 1)
        lmin = fminf(lmin, __shfl_xor(lmin, off, 32));
    if (lane == 0)
        atomicMin(minbits, __float_as_uint(lmin));  // positive floats: uint order == float order
}

__global__ void rl_finalize_loss(const unsigned int* __restrict__ minbits,
                                 float* __restrict__ out) {
    if (threadIdx.x == 0 && blockIdx.x == 0) {
        float d = __uint_as_float(*minbits);
        float l = 1.0f - d;                 // MARGIN_INTER = 1.0
        out[0] = (l > 0.0f ? l : 0.0f);     // * BETA (1.0)
    }
}

extern "C" void kernel_launch(void* const* d_in, const int* in_sizes, int n_in,
                              void* d_out, int out_size, void* d_ws, size_t ws_size,
                              hipStream_t stream) {
    const float* feats  = (const float*)d_in[0];
    const int*   labels = (const int*)d_in[1];

    float*        ws      = (float*)d_ws;
    float*        centers = ws;
    int*          counts  = (int*)(ws + OFF_COUNTS);
    float*        sq      = ws + OFF_SQ;
    unsigned int* minbits = (unsigned int*)(ws + OFF_MIN);
    int*          rowlist = (int*)(ws + OFF_ROWS);

    rl_init<<<(CPAD - NUM_CLASSES) * DDIM / 256, 256, 0, stream>>>(ws);
    rl_build_rows<<<NROWS / 256, 256, 0, stream>>>(labels, counts, rowlist);
    rl_class_reduce<<<NUM_CLASSES, 256, 0, stream>>>(feats, counts, rowlist, centers, sq);
    dim3 g(CPAD / 32, CPAD / 32);
    rl_gram_min<<<g, 32, 0, stream>>>(centers, sq, minbits);
    rl_finalize_loss<<<1, 64, 0, stream>>>(minbits, (float*)d_out);
}